// ResGCN_56487409877355
// MI455X (gfx1250) — compile-verified
//
#include <hip/hip_runtime.h>

#define NN 50000
#define NE 640000
#define NG 64
#define INF_ 8
#define H 128
#define H2 256
#define NC 4
#define NL 6
#define EPS_MSG 1e-7f
#define LN_EPS 1e-5f

typedef __attribute__((ext_vector_type(2))) float v2f;
typedef __attribute__((ext_vector_type(8))) float v8f;

// order-preserving float<->uint encoding for atomicMax on floats (0u == -inf)
__device__ __forceinline__ unsigned f2ord(float f) {
    unsigned b = __float_as_uint(f);
    return (b & 0x80000000u) ? ~b : (b | 0x80000000u);
}
__device__ __forceinline__ float ord2f(unsigned u) {
    return (u & 0x80000000u) ? __uint_as_float(u & 0x7fffffffu) : __uint_as_float(~u);
}

// ---------------- encoder: h = x @ enc_w + enc_b  (K=8, trivial) ----------------
__global__ void encoder_kernel(const float* __restrict__ x, const float* __restrict__ w,
                               const float* __restrict__ b, float* __restrict__ h) {
    int idx = blockIdx.x * blockDim.x + threadIdx.x;
    if (idx >= NN * H) return;
    int n = idx >> 7, j = idx & 127;
    const float* xr = x + n * INF_;
    float acc = b[j];
#pragma unroll
    for (int k = 0; k < INF_; ++k) acc += xr[k] * w[k * H + j];
    h[idx] = acc;
}

// ---------------- z = relu(LayerNorm(h)); one wave per node ----------------
__global__ void ln_relu_kernel(const float* __restrict__ h, const float* __restrict__ g,
                               const float* __restrict__ b, float* __restrict__ z) {
    int wid = (blockIdx.x * blockDim.x + threadIdx.x) >> 5;
    int lane = threadIdx.x & 31;
    if (wid >= NN) return;
    float4 v = ((const float4*)(h + (size_t)wid * H))[lane];
    float s = v.x + v.y + v.z + v.w;
    for (int off = 16; off; off >>= 1) s += __shfl_xor(s, off, 32);
    float mu = s * (1.0f / H);
    float dx = v.x - mu, dy = v.y - mu, dz = v.z - mu, dw = v.w - mu;
    float q = dx * dx + dy * dy + dz * dz + dw * dw;
    for (int off = 16; off; off >>= 1) q += __shfl_xor(q, off, 32);
    float rinv = rsqrtf(q * (1.0f / H) + LN_EPS);
    int c = lane * 4;
    float4 o;
    o.x = fmaxf(g[c + 0] * dx * rinv + b[c + 0], 0.f);
    o.y = fmaxf(g[c + 1] * dy * rinv + b[c + 1], 0.f);
    o.z = fmaxf(g[c + 2] * dz * rinv + b[c + 2], 0.f);
    o.w = fmaxf(g[c + 3] * dw * rinv + b[c + 3], 0.f);
    ((float4*)(z + (size_t)wid * H))[lane] = o;
}

// ---------------- edge pass 1: per-channel segment max of m*t ----------------
__global__ void edge_max_kernel(const float* __restrict__ z, const int* __restrict__ src,
                                const int* __restrict__ dst, const float* __restrict__ t_all,
                                int layer, unsigned* __restrict__ smax) {
    int e = (blockIdx.x * blockDim.x + threadIdx.x) >> 5;
    int lane = threadIdx.x & 31;
    if (e >= NE) return;
    float t = t_all[layer];
    int s = src[e], d = dst[e];
    float4 v = ((const float4*)(z + (size_t)s * H))[lane];
    unsigned* sp = smax + (size_t)d * H + lane * 4;
    atomicMax(sp + 0, f2ord((fmaxf(v.x, 0.f) + EPS_MSG) * t));
    atomicMax(sp + 1, f2ord((fmaxf(v.y, 0.f) + EPS_MSG) * t));
    atomicMax(sp + 2, f2ord((fmaxf(v.z, 0.f) + EPS_MSG) * t));
    atomicMax(sp + 3, f2ord((fmaxf(v.w, 0.f) + EPS_MSG) * t));
}

// ---------------- edge pass 2: denom += e, numer += m*e ----------------
__global__ void edge_sum_kernel(const float* __restrict__ z, const int* __restrict__ src,
                                const int* __restrict__ dst, const float* __restrict__ t_all,
                                int layer, const unsigned* __restrict__ smax,
                                float* __restrict__ denom, float* __restrict__ numer) {
    int e = (blockIdx.x * blockDim.x + threadIdx.x) >> 5;
    int lane = threadIdx.x & 31;
    if (e >= NE) return;
    float t = t_all[layer];
    int s = src[e], d = dst[e];
    float4 v = ((const float4*)(z + (size_t)s * H))[lane];
    size_t off = (size_t)d * H + lane * 4;
    float m0 = fmaxf(v.x, 0.f) + EPS_MSG;
    float m1 = fmaxf(v.y, 0.f) + EPS_MSG;
    float m2 = fmaxf(v.z, 0.f) + EPS_MSG;
    float m3 = fmaxf(v.w, 0.f) + EPS_MSG;
    float e0 = expf(m0 * t - ord2f(smax[off + 0]));
    float e1 = expf(m1 * t - ord2f(smax[off + 1]));
    float e2 = expf(m2 * t - ord2f(smax[off + 2]));
    float e3 = expf(m3 * t - ord2f(smax[off + 3]));
    atomicAdd(&denom[off + 0], e0);
    atomicAdd(&denom[off + 1], e1);
    atomicAdd(&denom[off + 2], e2);
    atomicAdd(&denom[off + 3], e3);
    atomicAdd(&numer[off + 0], m0 * e0);
    atomicAdd(&numer[off + 1], m1 * e1);
    atomicAdd(&numer[off + 2], m2 * e2);
    atomicAdd(&numer[off + 3], m3 * e3);
}

// ---------------- GENConv MLP, fp32 WMMA. 16 nodes per block, 4 waves ----------------
// h += (LN(relu) path): A = numer/denom + z; y1 = LN(A@w1 + b1); relu; h += y1@w2 + b2
__global__ __launch_bounds__(128) void mlp_kernel(
    const float* __restrict__ z, const float* __restrict__ numer, const float* __restrict__ denom,
    const float* __restrict__ w1, const float* __restrict__ b1,
    const float* __restrict__ mg, const float* __restrict__ mb,
    const float* __restrict__ w2, const float* __restrict__ b2, float* __restrict__ h) {
    __shared__ float sA[16][H + 2];   // stride 130 -> bank-conflict-free column reads
    __shared__ float sZ[16][H2 + 2];  // stride 258
    const int tid = threadIdx.x;
    const int wave = tid >> 5;
    const int lane = tid & 31;
    const int lo = lane & 15;
    const int hi = lane >> 4;  // lane-half selects K pair (f32 A/B frag layout)
    const int base = blockIdx.x * 16;

    // stage A = agg + z into LDS
    for (int idx = tid; idx < 16 * H; idx += 128) {
        int m = idx >> 7, k = idx & (H - 1);
        size_t off = (size_t)(base + m) * H + k;
        float de = denom[off];
        float agg = de > 0.f ? numer[off] / de : 0.f;
        sA[m][k] = agg + z[off];
    }
    __syncthreads();

    // GEMM1: [16x128] @ [128x256] + b1 -> sZ ; wave handles N-tiles {wave, wave+4, wave+8, wave+12}
    for (int ti = 0; ti < 4; ++ti) {
        int n0 = (wave + ti * 4) * 16;
        float bias = b1[n0 + lo];
        v8f c;
#pragma unroll
        for (int j = 0; j < 8; ++j) c[j] = bias;
        for (int k0 = 0; k0 < H; k0 += 4) {
            int kk = k0 + (hi << 1);
            v2f a, bf;
            a.x = sA[lo][kk];
            a.y = sA[lo][kk + 1];
            bf.x = w1[kk * H2 + n0 + lo];
            bf.y = w1[(kk + 1) * H2 + n0 + lo];
            c = __builtin_amdgcn_wmma_f32_16x16x4_f32(false, a, false, bf, (short)0, c, false, false);
        }
#pragma unroll
        for (int j = 0; j < 8; ++j) sZ[j + 8 * hi][n0 + lo] = c[j];
    }
    __syncthreads();

    // LayerNorm(256) + ReLU in LDS: 8 lanes per node
    {
        int node = tid >> 3;
        int c0 = (tid & 7) * 32;
        float s = 0.f, q = 0.f;
#pragma unroll
        for (int i = 0; i < 32; ++i) s += sZ[node][c0 + i];
        s += __shfl_xor(s, 1, 32); s += __shfl_xor(s, 2, 32); s += __shfl_xor(s, 4, 32);
        float mu = s * (1.0f / H2);
#pragma unroll
        for (int i = 0; i < 32; ++i) { float d = sZ[node][c0 + i] - mu; q += d * d; }
        q += __shfl_xor(q, 1, 32); q += __shfl_xor(q, 2, 32); q += __shfl_xor(q, 4, 32);
        float rinv = rsqrtf(q * (1.0f / H2) + LN_EPS);
#pragma unroll
        for (int i = 0; i < 32; ++i) {
            int c = c0 + i;
            float v = (sZ[node][c] - mu) * rinv;
            sZ[node][c] = fmaxf(mg[c] * v + mb[c], 0.f);
        }
    }
    __syncthreads();

    // GEMM2: [16x256] @ [256x128] + b2 ; residual h += result
    for (int ti = 0; ti < 2; ++ti) {
        int n0 = (wave + ti * 4) * 16;
        float bias = b2[n0 + lo];
        v8f c;
#pragma unroll
        for (int j = 0; j < 8; ++j) c[j] = bias;
        for (int k0 = 0; k0 < H2; k0 += 4) {
            int kk = k0 + (hi << 1);
            v2f a, bf;
            a.x = sZ[lo][kk];
            a.y = sZ[lo][kk + 1];
            bf.x = w2[kk * H + n0 + lo];
            bf.y = w2[(kk + 1) * H + n0 + lo];
            c = __builtin_amdgcn_wmma_f32_16x16x4_f32(false, a, false, bf, (short)0, c, false, false);
        }
#pragma unroll
        for (int j = 0; j < 8; ++j) {
            size_t off = (size_t)(base + j + 8 * hi) * H + n0 + lo;
            h[off] += c[j];
        }
    }
}

// ---------------- global add pool + classifier ----------------
__global__ void pool_kernel(const float* __restrict__ h, const int* __restrict__ batch,
                            float* __restrict__ pooled) {
    int wid = (blockIdx.x * blockDim.x + threadIdx.x) >> 5;
    int lane = threadIdx.x & 31;
    if (wid >= NN) return;
    float4 v = ((const float4*)(h + (size_t)wid * H))[lane];
    float* p = pooled + (size_t)batch[wid] * H + lane * 4;
    atomicAdd(p + 0, v.x);
    atomicAdd(p + 1, v.y);
    atomicAdd(p + 2, v.z);
    atomicAdd(p + 3, v.w);
}

__global__ void cls_kernel(const float* __restrict__ pooled, const float* __restrict__ w,
                           const float* __restrict__ b, float* __restrict__ out) {
    int idx = threadIdx.x;  // 256 = 64 graphs x 4 classes
    int g = idx >> 2, c = idx & 3;
    float acc = b[c];
#pragma unroll 8
    for (int k = 0; k < H; ++k) acc += pooled[g * H + k] * w[k * NC + c];
    out[idx] = acc;
}

extern "C" void kernel_launch(void* const* d_in, const int* in_sizes, int n_in,
                              void* d_out, int out_size, void* d_ws, size_t ws_size,
                              hipStream_t stream) {
    const float* x     = (const float*)d_in[0];
    const int*   ei    = (const int*)d_in[1];
    const int*   batch = (const int*)d_in[2];
    const float* enc_w = (const float*)d_in[3];
    const float* enc_b = (const float*)d_in[4];
    const float* ln_g  = (const float*)d_in[5];
    const float* ln_b  = (const float*)d_in[6];
    const float* t_all = (const float*)d_in[7];
    const float* w1    = (const float*)d_in[8];
    const float* b1    = (const float*)d_in[9];
    const float* mg    = (const float*)d_in[10];
    const float* mb    = (const float*)d_in[11];
    const float* w2    = (const float*)d_in[12];
    const float* b2    = (const float*)d_in[13];
    const float* cls_w = (const float*)d_in[14];
    const float* cls_b = (const float*)d_in[15];

    const int* src = ei;             // edge_index[0]
    const int* dst = ei + NE;        // edge_index[1]

    const size_t NF = (size_t)NN * H;  // 6.4M floats per node-feature buffer
    float*    h      = (float*)d_ws;
    float*    z      = h + NF;
    unsigned* smax   = (unsigned*)(z + NF);
    float*    denom  = (float*)(smax + NF);
    float*    numer  = denom + NF;
    float*    pooled = numer + NF;

    encoder_kernel<<<(NN * H + 255) / 256, 256, 0, stream>>>(x, enc_w, enc_b, h);

    for (int l = 0; l < NL; ++l) {
        ln_relu_kernel<<<(NN * 32 + 255) / 256, 256, 0, stream>>>(h, ln_g + l * H, ln_b + l * H, z);
        hipMemsetAsync(smax, 0, NF * 4, stream);   // 0u encodes -inf
        hipMemsetAsync(denom, 0, NF * 4, stream);
        hipMemsetAsync(numer, 0, NF * 4, stream);
        edge_max_kernel<<<(NE * 32) / 256, 256, 0, stream>>>(z, src, dst, t_all, l, smax);
        edge_sum_kernel<<<(NE * 32) / 256, 256, 0, stream>>>(z, src, dst, t_all, l, smax, denom, numer);
        mlp_kernel<<<NN / 16, 128, 0, stream>>>(z, numer, denom,
                                                w1 + (size_t)l * H * H2, b1 + l * H2,
                                                mg + l * H2, mb + l * H2,
                                                w2 + (size_t)l * H2 * H, b2 + l * H, h);
    }

    hipMemsetAsync(pooled, 0, (size_t)NG * H * 4, stream);
    pool_kernel<<<(NN * 32 + 255) / 256, 256, 0, stream>>>(h, batch, pooled);
    cls_kernel<<<1, 256, 0, stream>>>(pooled, cls_w, cls_b, (float*)d_out);
}